// MyLoss_51204600103232
// MI455X (gfx1250) — compile-verified
//
#include <hip/hip_runtime.h>
#include <math.h>

typedef __attribute__((ext_vector_type(2))) float v2f;
typedef __attribute__((ext_vector_type(8))) float v8f;

#define DIMS    768
#define WAVES   8
#define KCHUNK  (DIMS / WAVES)   // 96 columns per wave
#define KSTEPS  (KCHUNK / 4)     // 24 WMMA K=4 steps per wave
#define MARGIN  0.2f

// One block = one 16-row tile. 8 waves split the K dimension.
// Per K=4 step, lane L supplies float2 X[row = L%16][kbase + 2*(L/16) + {0,1}],
// which is simultaneously the A-operand (16x4 MxK) and B-operand (4x16 KxN,
// B[k][n] = X[n][k]) layout for V_WMMA_F32_16X16X4_F32. diag(A*B) then gives
// 16 row-dots per accumulator.
__global__ __launch_bounds__(256)
void diag_margin_wmma(const float* __restrict__ img,
                      const float* __restrict__ pos,
                      const float* __restrict__ neg,
                      float* __restrict__ out) {
    const int lane = threadIdx.x & 31;
    const int wave = threadIdx.x >> 5;
    const int mrow = lane & 15;   // which of the 16 rows this lane covers
    const int kh   = lane >> 4;   // K-half within the 4-wide step
    const int tile = blockIdx.x;
    const int row  = tile * 16 + mrow;

    const size_t off = (size_t)row * DIMS + (size_t)wave * KCHUNK + 2 * kh;
    const v2f* ip = (const v2f*)(img + off);
    const v2f* pp = (const v2f*)(pos + off);
    const v2f* np = (const v2f*)(neg + off);

    v8f a_ip = {};  // img . pos
    v8f a_pp = {};  // pos . pos
    v8f a_in = {};  // img . neg
    v8f a_nn = {};  // neg . neg

#pragma unroll 4
    for (int s = 0; s < KSTEPS; ++s) {
        v2f vi = ip[2 * s];   // step advances K by 4 floats = 2 v2f
        v2f vp = pp[2 * s];
        v2f vn = np[2 * s];
        a_ip = __builtin_amdgcn_wmma_f32_16x16x4_f32(false, vi, false, vp,
                                                     (short)0, a_ip, false, false);
        a_pp = __builtin_amdgcn_wmma_f32_16x16x4_f32(false, vp, false, vp,
                                                     (short)0, a_pp, false, false);
        a_in = __builtin_amdgcn_wmma_f32_16x16x4_f32(false, vi, false, vn,
                                                     (short)0, a_in, false, false);
        a_nn = __builtin_amdgcn_wmma_f32_16x16x4_f32(false, vn, false, vn,
                                                     (short)0, a_nn, false, false);
    }

    // Extract diag of the 16x16 f32 C/D layout:
    //   VGPR r, lane r      -> element (M=r,   N=r)
    //   VGPR r, lane r+24   -> element (M=r+8, N=r+8)
    float d_ip = 0.f, d_pp = 0.f, d_in = 0.f, d_nn = 0.f;
#pragma unroll
    for (int r = 0; r < 8; ++r) {
        float t, lo, hi;
        t = a_ip[r]; lo = __shfl(t, r, 32); hi = __shfl(t, r + 24, 32);
        if (lane == r) d_ip = lo; if (lane == r + 8) d_ip = hi;
        t = a_pp[r]; lo = __shfl(t, r, 32); hi = __shfl(t, r + 24, 32);
        if (lane == r) d_pp = lo; if (lane == r + 8) d_pp = hi;
        t = a_in[r]; lo = __shfl(t, r, 32); hi = __shfl(t, r + 24, 32);
        if (lane == r) d_in = lo; if (lane == r + 8) d_in = hi;
        t = a_nn[r]; lo = __shfl(t, r, 32); hi = __shfl(t, r + 24, 32);
        if (lane == r) d_nn = lo; if (lane == r + 8) d_nn = hi;
    }

    // Combine partial K-chunks across the 8 waves.
    __shared__ float part[WAVES][4][16];
    if (lane < 16) {
        part[wave][0][mrow] = d_ip;
        part[wave][1][mrow] = d_pp;
        part[wave][2][mrow] = d_in;
        part[wave][3][mrow] = d_nn;
    }
    __syncthreads();

    if (threadIdx.x < 16) {
        float s_ip = 0.f, s_pp = 0.f, s_in = 0.f, s_nn = 0.f;
#pragma unroll
        for (int w = 0; w < WAVES; ++w) {
            s_ip += part[w][0][threadIdx.x];
            s_pp += part[w][1][threadIdx.x];
            s_in += part[w][2][threadIdx.x];
            s_nn += part[w][3][threadIdx.x];
        }
        const float pos_score = s_ip / sqrtf(s_pp);
        const float neg_score = s_in / sqrtf(s_nn);
        const float v = neg_score - pos_score + MARGIN;
        out[tile * 16 + threadIdx.x] = v > 0.f ? v : 0.f;
    }
}

extern "C" void kernel_launch(void* const* d_in, const int* in_sizes, int n_in,
                              void* d_out, int out_size, void* d_ws, size_t ws_size,
                              hipStream_t stream) {
    const float* img = (const float*)d_in[0];  // image_feat   [N, 768]
    const float* pos = (const float*)d_in[1];  // pos_GCN_emb  [N, 768]
    const float* neg = (const float*)d_in[2];  // neg_GCN_emb  [N, 768]
    // d_in[3], d_in[4] (text feats) are multiplied by zero in the reference:
    // never read them -> saves 50 MB of HBM traffic.
    float* out = (float*)d_out;

    const int n_rows = in_sizes[0] / DIMS;     // 8192
    const int tiles  = n_rows / 16;            // 512 blocks

    diag_margin_wmma<<<tiles, 256, 0, stream>>>(img, pos, neg, out);
}